// Involution_56152402428073
// MI455X (gfx1250) — compile-verified
//
#include <hip/hip_runtime.h>

// ---------------------------------------------------------------------------
// Involution (b=4, C=256, 56x56, G=16, K=7) for MI455X / gfx1250, wave32.
//   Kernel 1: r = relu(BN(x @ w1^T + b1))      GEMM M=12544 K=256 N=64 (WMMA f32)
//   Kernel 2: wm = r @ w2^T + b2 (-> LDS)      GEMM M=16 K=64 N=784 per block (WMMA f32)
//             out = sum_k wm[g,k] * unfold(x)  per-pixel 7x7 gather (L2-resident)
// Entirely fp32: the problem is ~2 GFLOP vs a ~2-3us HBM floor at 23.3 TB/s,
// so we keep full precision and route GEMMs through V_WMMA_F32_16X16X4_F32.
// ---------------------------------------------------------------------------

typedef __attribute__((ext_vector_type(2))) float v2f;
typedef __attribute__((ext_vector_type(8))) float v8f;

#define BATCH 4
#define CIN   256
#define HH    56
#define WW    56
#define HW    (HH * WW)            // 3136
#define CHW   (CIN * HW)           // 802816
#define NPIX  (BATCH * HW)         // 12544
#define CR    64                   // reduced channels
#define GRP   16
#define GC    16
#define KK    49                   // 7x7
#define NOUT  (GRP * KK)           // 784

// ---------------------------------------------------------------------------
// Kernel 1: reduce 1x1 conv + BN(inference) + ReLU, fused.
// Grid: 392 blocks x 256 threads (8 waves). Wave w: pixels p0+16*(w>>2)..+15,
// output channels 16*(w&3)..+15. K loop: 256 in steps of 4 -> 64 WMMAs/wave.
// ---------------------------------------------------------------------------
__global__ __launch_bounds__(256) void involution_reduce_gemm(
    const float* __restrict__ xg,   // (B, C, H, W)
    const float* __restrict__ w1,   // (CR, C) row-major
    const float* __restrict__ b1,
    const float* __restrict__ gamma,
    const float* __restrict__ beta,
    const float* __restrict__ rmean,
    const float* __restrict__ rvar,
    float* __restrict__ r)          // (NPIX, CR) row-major
{
    const int t    = threadIdx.x;
    const int wave = t >> 5;
    const int lane = t & 31;
    const int lm   = lane & 15;     // A: M index / B: N index / C: N index
    const int lh   = lane >> 4;     // K half selector

    const int p0 = blockIdx.x * 32 + (wave >> 2) * 16;  // pixel tile base
    const int nb = (wave & 3) * 16;                     // output-channel tile base

    // A-row pixel for this lane
    const int  p      = p0 + lm;
    const int  bb     = p / HW;
    const int  rem    = p - bb * HW;
    const long pixoff = (long)bb * CHW + rem;           // + k*HW per channel k

    // BN parameters for this lane's output channel (N = lane%16, same for all
    // 8 accumulator VGPRs per the C/D layout).
    const int   n     = nb + lm;
    const float bias  = b1[n];
    const float mean  = rmean[n];
    const float scale = gamma[n] * rsqrtf(rvar[n] + 1e-5f);
    const float shift = beta[n];

    const float* brow = w1 + n * CIN;   // w1 row for N=lm (K-contiguous)

    v8f acc = {};
    for (int k0 = 0; k0 < CIN; k0 += 4) {
        const int ka = k0 + 2 * lh;
        // A fragment (16x4): lane holds (M=lm, K=ka) and (M=lm, K=ka+1);
        // channel stride in x is HW -> two strided dword loads.
        v2f a;
        a.x = xg[pixoff + (long)ka * HW];
        a.y = xg[pixoff + (long)(ka + 1) * HW];
        // B fragment (4x16): lane holds (K=ka, N=lm), (K=ka+1, N=lm);
        // contiguous in w1 row, 8B aligned (ka even).
        v2f b = *(const v2f*)(brow + ka);
        acc = __builtin_amdgcn_wmma_f32_16x16x4_f32(
            false, a, false, b, (short)0, acc, false, false);
    }

    // C/D layout: VGPR v, lane -> (M = v + 8*lh, N = lm). Apply BN+ReLU, store
    // pixel-major so kernel 2's A loads are contiguous float2s.
#pragma unroll
    for (int v = 0; v < 8; ++v) {
        const int m = v + 8 * lh;
        float val = (acc[v] + bias - mean) * scale + shift;
        val = fmaxf(val, 0.0f);
        r[(p0 + m) * CR + n] = val;
    }
}

// ---------------------------------------------------------------------------
// Kernel 2: span 1x1 conv into LDS + per-pixel 7x7 weighted gather, fused.
// Grid: 784 blocks x 256 threads; one block = 16 pixels.
// Phase A: 8 waves split 49 N-tiles of the 16x784 wm tile (K=64 -> 16 WMMAs);
//          A fragment preloaded once per wave (16 v2f regs), reused per tile.
// Phase B: thread (pi = t&15, g = t>>4) computes 16 output channels; pixel in
//          low lane bits keeps the x gather coalesced; x stays hot in L2.
// ---------------------------------------------------------------------------
__global__ __launch_bounds__(256) void involution_span_apply(
    const float* __restrict__ xg,   // (B, C, H, W)
    const float* __restrict__ r,    // (NPIX, CR)
    const float* __restrict__ w2,   // (NOUT, CR) row-major
    const float* __restrict__ b2,   // (NOUT,)
    float* __restrict__ out)        // (B, C, H, W)
{
    __shared__ float wm[16 * NOUT];  // 50176 B (<< 320 KB/WGP)

    const int t    = threadIdx.x;
    const int wave = t >> 5;
    const int lane = t & 31;
    const int lm   = lane & 15;
    const int lh   = lane >> 4;
    const int p0   = blockIdx.x * 16;

    // Preload this wave's A fragments for all 16 K-steps (K=64).
    // areg[i] holds (M=lm, K=4i+2*lh) and (M=lm, K=4i+2*lh+1): contiguous.
    v2f areg[16];
    const float* arow = r + (p0 + lm) * CR + 2 * lh;
#pragma unroll
    for (int i = 0; i < 16; ++i) areg[i] = *(const v2f*)(arow + 4 * i);

    for (int nt = wave; nt < KK; nt += 8) {
        const int nb = nt * 16;
        const int n  = nb + lm;
        const float* brow = w2 + n * CR + 2 * lh;
        v8f acc = {};
#pragma unroll
        for (int i = 0; i < 16; ++i) {
            v2f b = *(const v2f*)(brow + 4 * i);
            acc = __builtin_amdgcn_wmma_f32_16x16x4_f32(
                false, areg[i], false, b, (short)0, acc, false, false);
        }
        const float bias = b2[n];
#pragma unroll
        for (int v = 0; v < 8; ++v)
            wm[(v + 8 * lh) * NOUT + n] = acc[v] + bias;
    }

    __syncthreads();

    // ---- Involution apply -------------------------------------------------
    const int pi = t & 15;          // pixel within tile (low bits -> coalesced)
    const int g  = t >> 4;          // group 0..15
    const int p  = p0 + pi;
    const int bb = p / HW;
    const int rm = p - bb * HW;
    const int yy = rm / WW;
    const int xx = rm - yy * WW;

    const float* wp    = wm + pi * NOUT + g * KK;
    const float* xbase = xg + (long)bb * CHW + (long)(g * GC) * HW;

    float acc2[GC];
#pragma unroll
    for (int cc = 0; cc < GC; ++cc) acc2[cc] = 0.0f;

    for (int k = 0; k < KK; ++k) {
        const int dy = k / 7 - 3;
        const int dx = k - (k / 7) * 7 - 3;
        const int y2 = yy + dy;
        const int x2 = xx + dx;
        if ((unsigned)y2 < (unsigned)HH && (unsigned)x2 < (unsigned)WW) {
            const float wk = wp[k];
            const float* xp = xbase + y2 * WW + x2;
#pragma unroll
            for (int cc = 0; cc < GC; ++cc)
                acc2[cc] = fmaf(wk, xp[cc * HW], acc2[cc]);
        }
    }

    float* op = out + (long)bb * CHW + (long)(g * GC) * HW + yy * WW + xx;
#pragma unroll
    for (int cc = 0; cc < GC; ++cc) op[cc * HW] = acc2[cc];
}

// ---------------------------------------------------------------------------
extern "C" void kernel_launch(void* const* d_in, const int* in_sizes, int n_in,
                              void* d_out, int out_size, void* d_ws, size_t ws_size,
                              hipStream_t stream) {
    const float* x     = (const float*)d_in[0];
    const float* w1    = (const float*)d_in[1];
    const float* b1    = (const float*)d_in[2];
    const float* gamma = (const float*)d_in[3];
    const float* beta  = (const float*)d_in[4];
    const float* rmean = (const float*)d_in[5];
    const float* rvar  = (const float*)d_in[6];
    const float* w2    = (const float*)d_in[7];
    const float* b2    = (const float*)d_in[8];

    float* r = (float*)d_ws;   // NPIX * CR floats = 3.06 MB scratch

    involution_reduce_gemm<<<NPIX / 32, 256, 0, stream>>>(
        x, w1, b1, gamma, beta, rmean, rvar, r);
    involution_span_apply<<<NPIX / 16, 256, 0, stream>>>(
        x, r, w2, b2, (float*)d_out);
}